// Conv1d_NN_1494648619740
// MI455X (gfx1250) — compile-verified
//
#include <hip/hip_runtime.h>

// CDNA5 / gfx1250, wave32. Fused NN-conv1d:
//   per-batch token Gram tiles via V_WMMA_F32_16X16X4_F32 (fp32 kept: top-k
//   ordering is precision sensitive), fused top-8 selection, then the
//   gathered stride-K conv as a second fp32 WMMA GEMM.
// j-tiles are streamed into LDS with the CDNA5 async global->LDS path
// (ASYNCcnt double buffering) when the toolchain exposes the builtins.

typedef __attribute__((ext_vector_type(2))) float v2f;
typedef __attribute__((ext_vector_type(8))) float v8f;
typedef __attribute__((ext_vector_type(4))) int   v4i;

typedef __attribute__((address_space(1))) v4i* gv4i_p;   // global 128b chunk
typedef __attribute__((address_space(3))) v4i* lv4i_p;   // LDS 128b chunk

#define B_    16
#define C_    64
#define T_    2048
#define K_    8
#define OC_   64
#define TILE  16
#define NJT   (T_ / TILE)     // 128 j tiles
#define WAVES 4
#define BLOCK (WAVES * 32)
#define CK    (C_ * K_)       // 512 reduction dim of the conv GEMM

#if __has_builtin(__builtin_amdgcn_global_load_async_to_lds_b128) && \
    __has_builtin(__builtin_amdgcn_s_wait_asynccnt)
#define USE_ASYNC 1
#else
#define USE_ASYNC 0
#endif

#if USE_ASYNC
#define ASYNC_WAIT(n) __builtin_amdgcn_s_wait_asynccnt(n)
#else
#define ASYNC_WAIT(n) ((void)0)
#endif

// Stage one 64(ch) x 16(token) j-tile into LDS, channel-major [C_][TILE].
// Each of the 8 transfers moves 8 rows x 64B (32 lanes x 16B).
__device__ __forceinline__ void stage_tile(const float* __restrict__ xb,
                                           float* dst, int j0, int lane)
{
    const int crow = lane >> 2;   // 0..7 : channel row within group of 8
    const int quad = lane & 3;    // 0..3 : 16B chunk within the 64B row
#pragma unroll
    for (int i = 0; i < 8; ++i) {
        const int c = 8 * i + crow;
        const float* g = xb + c * T_ + j0 + quad * 4;
        float*       l = dst + c * TILE + quad * 4;
#if USE_ASYNC
        __builtin_amdgcn_global_load_async_to_lds_b128(
            (gv4i_p)g, (lv4i_p)l, 0, 0);
#else
        *(float4*)l = *(const float4*)g;   // sync fallback
#endif
    }
}

__launch_bounds__(BLOCK)
__global__ void conv1d_nn_kernel(const float* __restrict__ x,
                                 const float* __restrict__ w,
                                 const float* __restrict__ bias,
                                 float* __restrict__ out)
{
    // ---- LDS (~50 KB of 320 KB/WGP; jbuf unioned with phase-2 buffers) ----
    __shared__ float xiT[C_][TILE];                 // A tile, channel-major
    __shared__ float ni[TILE];                      // row norms
    __shared__ float distb[WAVES][TILE][TILE + 1];  // per-wave dist tile, padded
    __shared__ float cd[WAVES][TILE][K_];           // per-wave top-8 dists
    __shared__ int   ci[WAVES][TILE][K_];           // per-wave top-8 indices
    __shared__ int   idxF[TILE][K_];                // merged neighbor indices
    __shared__ union {
        float jbuf[WAVES][2][C_][TILE];             // phase-1 j-tile double buffers
        struct {
            float prime[TILE][CK + 1];              // gathered features, padded
            float otile[TILE][OC_];                 // output transpose buffer
        } p2;
    } u;

    const int tid  = threadIdx.x;
    const int wv   = tid >> 5;
    const int lane = tid & 31;
    const int col  = lane & 15;   // N (or M for A-frag rows)
    const int half = lane >> 4;   // upper/lower 16 lanes

    const int blk   = blockIdx.x;
    const int b     = blk / NJT;          // batch
    const int itile = blk % NJT;          // 16-token row tile
    const int i0    = itile * TILE;

    const float* xb = x + (size_t)b * C_ * T_;

    // warm L2 for this wave's weight rows (uniform -> no exec-mask branch)
    __builtin_prefetch(w + (wv * TILE + col) * CK, 0, 0);

    // ---- Phase 0: stage A tile x[b, :, i0:i0+16] channel-major + row norms ----
    for (int e = tid; e < C_ * TILE; e += BLOCK) {
        const int c = e >> 4, m = e & 15;
        xiT[c][m] = xb[c * T_ + i0 + m];
    }
    __syncthreads();
    if (tid < TILE) {
        float s = 0.f;
        for (int c = 0; c < C_; ++c) { const float v = xiT[c][tid]; s += v * v; }
        ni[tid] = s;
    }
    __syncthreads();

    // ---- Phase 1: distance tiles (WMMA fp32) + running top-8 per row ----
    float d8[K_]; int i8[K_];
#pragma unroll
    for (int q = 0; q < K_; ++q) { d8[q] = 3.4e38f; i8[q] = 0; }

    // prologue: stream this wave's first j-tile into buffer 0
    stage_tile(xb, &u.jbuf[wv][0][0][0], wv * TILE, lane);

    int p = 0;
    for (int jt = wv; jt < NJT; jt += WAVES, p ^= 1) {
        const int j0 = jt * TILE;

        // stream the next tile into the other buffer, then wait for this one
        if (jt + WAVES < NJT) {
            stage_tile(xb, &u.jbuf[wv][p ^ 1][0][0], (jt + WAVES) * TILE, lane);
            ASYNC_WAIT(8);    // allow the 8 just-issued; this tile's 8 are done
        } else {
            ASYNC_WAIT(0);    // drain: last tile must be resident
        }
        const float* jb = &u.jbuf[wv][p][0][0];

        v8f acc = {};
        float sq = 0.f;
#pragma unroll
        for (int kk = 0; kk < C_ / 4; ++kk) {       // K=64 in steps of 4
            const int c0 = 4 * kk + 2 * half;       // lane's k-slice within group
            v2f af, bf;
            af.x = xiT[c0    ][col];                // A[m=col][k] (hoisted to regs)
            af.y = xiT[c0 + 1][col];
            bf.x = jb[(c0    ) * TILE + col];       // B[k][n=col] from LDS
            bf.y = jb[(c0 + 1) * TILE + col];
            sq += bf.x * bf.x + bf.y * bf.y;        // column-norm partial (free)
            acc = __builtin_amdgcn_wmma_f32_16x16x4_f32(
                      false, af, false, bf, (short)0, acc, false, false);
        }
        // column norm: combine the two lane-halves (each covered half the channels)
        const float nj = sq + __shfl_xor(sq, 16, 32);

        // dist = max(ni + nj - 2*dot, 0) -> padded LDS tile
#pragma unroll
        for (int r = 0; r < 8; ++r) {
            const int m = r + 8 * half;             // C/D layout: M = r + 8*half
            const float d = ni[m] + nj - 2.f * acc[r];
            distb[wv][m][col] = d > 0.f ? d : 0.f;
        }
        // scan 16 columns of this tile into the sorted top-8 (row = lane)
        if (lane < 16) {
#pragma unroll
            for (int c = 0; c < TILE; ++c) {
                const float d = distb[wv][lane][c];
                if (d < d8[K_ - 1]) {
                    d8[K_ - 1] = d; i8[K_ - 1] = j0 + c;
#pragma unroll
                    for (int q = K_ - 1; q > 0; --q) {
                        if (d8[q] < d8[q - 1]) {
                            const float td = d8[q]; d8[q] = d8[q - 1]; d8[q - 1] = td;
                            const int   ti = i8[q]; i8[q] = i8[q - 1]; i8[q - 1] = ti;
                        }
                    }
                }
            }
        }
    }

    // ---- merge the 4 per-wave top-8 lists per row ----
    if (lane < 16) {
#pragma unroll
        for (int q = 0; q < K_; ++q) { cd[wv][lane][q] = d8[q]; ci[wv][lane][q] = i8[q]; }
    }
    __syncthreads();
    if (wv == 0 && lane < 16) {
        float f8[K_]; int fi[K_];
#pragma unroll
        for (int q = 0; q < K_; ++q) { f8[q] = 3.4e38f; fi[q] = 0; }
        for (int ws = 0; ws < WAVES; ++ws) {
#pragma unroll
            for (int q = 0; q < K_; ++q) {
                const float d = cd[ws][lane][q]; const int j = ci[ws][lane][q];
                if (d < f8[K_ - 1]) {
                    f8[K_ - 1] = d; fi[K_ - 1] = j;
#pragma unroll
                    for (int pq = K_ - 1; pq > 0; --pq) {
                        if (f8[pq] < f8[pq - 1]) {
                            const float td = f8[pq]; f8[pq] = f8[pq - 1]; f8[pq - 1] = td;
                            const int   ti = fi[pq]; fi[pq] = fi[pq - 1]; fi[pq - 1] = ti;
                        }
                    }
                }
            }
        }
#pragma unroll
        for (int q = 0; q < K_; ++q) idxF[lane][q] = fi[q];
    }
    __syncthreads();   // also retires jbuf before the union's phase-2 view

    // ---- Phase 2: gather prime[m][ck], ck = c*8 + k (matches w[o,c,k] flat) ----
    for (int e = tid; e < TILE * CK; e += BLOCK) {
        const int m = e >> 9, ck = e & (CK - 1);
        const int c = ck >> 3, kq = ck & 7;
        u.p2.prime[m][ck] = xb[c * T_ + idxF[m][kq]];   // L2-resident gather
    }
    __syncthreads();

    // conv GEMM: out(16 tokens x 16 outputs) per wave, K = 512 in steps of 4,
    // 1-deep software pipeline so w/prime loads overlap the WMMA chain.
    const int o0 = wv * TILE;
    const float* wrow = w + (size_t)(o0 + col) * CK;
    const float* prow = &u.p2.prime[col][0];

    v8f acc2 = {};
    v2f af0, bf0, af1, bf1;
    {
        const int ck0 = 2 * half;
        af0.x = prow[ck0]; af0.y = prow[ck0 + 1];
        bf0.x = wrow[ck0]; bf0.y = wrow[ck0 + 1];
    }
#pragma unroll 8
    for (int s = 0; s < CK / 4; ++s) {
        if (s + 1 < CK / 4) {
            const int nk = 4 * (s + 1) + 2 * half;
            af1.x = prow[nk]; af1.y = prow[nk + 1];
            bf1.x = wrow[nk]; bf1.y = wrow[nk + 1];
        }
        acc2 = __builtin_amdgcn_wmma_f32_16x16x4_f32(
                   false, af0, false, bf0, (short)0, acc2, false, false);
        af0 = af1; bf0 = bf1;
    }
    const float bo = bias[o0 + col];
#pragma unroll
    for (int r = 0; r < 8; ++r) {
        const int m = r + 8 * half;
        u.p2.otile[m][o0 + col] = acc2[r] + bo;
    }
    __syncthreads();

    // coalesced store: out[b, o, i0+m]
    float* ob = out + (size_t)b * OC_ * T_;
    for (int e = tid; e < TILE * OC_; e += BLOCK) {
        const int o = e >> 4, m = e & 15;
        ob[o * T_ + i0 + m] = u.p2.otile[m][o];
    }
}

extern "C" void kernel_launch(void* const* d_in, const int* in_sizes, int n_in,
                              void* d_out, int out_size, void* d_ws, size_t ws_size,
                              hipStream_t stream) {
    (void)in_sizes; (void)n_in; (void)out_size; (void)d_ws; (void)ws_size;
    const float* x    = (const float*)d_in[0];   // [16, 64, 2048]
    const float* w    = (const float*)d_in[1];   // [64, 64, 8]
    const float* bias = (const float*)d_in[2];   // [64]
    float* out        = (float*)d_out;           // [16, 64, 2048]

    dim3 grid(B_ * (T_ / TILE));   // 2048 workgroups
    dim3 block(BLOCK);             // 128 threads = 4 wave32
    conv1d_nn_kernel<<<grid, block, 0, stream>>>(x, w, bias, out);
}